// TritonLinearETRow_34626026340512
// MI455X (gfx1250) — compile-verified
//
#include <hip/hip_runtime.h>

typedef __attribute__((ext_vector_type(2))) float v2f;
typedef __attribute__((ext_vector_type(8))) float v8f;

#define DIN   4096
#define DOUT  4096
#define K0_   16

// Kernel 1: c = sum_{j in [16,4096)} W[0,j] * x2[0,j]  (single block reduction)
__global__ __launch_bounds__(256) void et_reduce_c(const float* __restrict__ x,
                                                   const float* __restrict__ W,
                                                   float* __restrict__ c_out) {
  __shared__ float sm[256];
  const int t = threadIdx.x;
  float p = 0.0f;
  for (int j = K0_ + t; j < DOUT; j += 256) p += W[j] * x[j];
  sm[t] = p;
  __syncthreads();
  for (int s = 128; s > 0; s >>= 1) {
    if (t < s) sm[t] += sm[t + s];
    __syncthreads();
  }
  if (t == 0) c_out[0] = sm[0];
}

// Kernel 2: out[m,n] = y1[m,n] + c with y1 = xk @ wk^T (K=16).
// (passed == False identically: tstat >= 0 or NaN, TAU < 0, so the reference's
//  where()/active path reduces exactly to y1 + c for every element.)
//
// One wave -> 16 (M) x 64 (N) outputs: A tile (16x16 f32) held in 8 VGPRs and
// reused across 4 N-tiles; per N-tile 4 chained v_wmma_f32_16x16x4_f32.
// 8 waves/block -> 16 x 512 per block.
__global__ __launch_bounds__(256) void et_rank16_wmma(const float* __restrict__ x,
                                                      const float* __restrict__ W,
                                                      const float* __restrict__ cptr,
                                                      float* __restrict__ out) {
  const int lane = threadIdx.x & 31;
  const int wave = threadIdx.x >> 5;
  const int half = lane >> 4;   // 0: lanes 0-15, 1: lanes 16-31
  const int l15  = lane & 15;

  const int m_base = blockIdx.y * 16;
  const int n_base = (blockIdx.x * 8 + wave) * 64;

  const float cval = cptr[0];

  // A (16x4 f32 per K-chunk): lane holds row (m_base + l15),
  // K = 4*kk + 2*half + {0,1}  -> one b64 load per chunk, 4 chunks total.
  const float* ap = x + (size_t)(m_base + l15) * DIN + 2 * half;
  v2f a[4];
#pragma unroll
  for (int kk = 0; kk < 4; ++kk) a[kk] = *(const v2f*)(ap + 4 * kk);

#pragma unroll
  for (int nt = 0; nt < 4; ++nt) {
    // B (4x16 per K-chunk): B[k,n] = W[n,k]; lane holds col (n0 + l15).
    const int n0 = n_base + nt * 16;
    const float* bp = W + (size_t)(n0 + l15) * DIN + 2 * half;

    v8f acc = {};
#pragma unroll
    for (int kk = 0; kk < 4; ++kk) {
      v2f b = *(const v2f*)(bp + 4 * kk);
      acc = __builtin_amdgcn_wmma_f32_16x16x4_f32(false, a[kk], false, b, (short)0,
                                                  acc, false, false);
    }

    // C/D layout: VGPR i -> row (m_base + i + 8*half), col (n0 + l15).
    float* op = out + (size_t)(m_base + 8 * half) * DOUT + (size_t)(n0 + l15);
#pragma unroll
    for (int i = 0; i < 8; ++i) {
      __builtin_nontemporal_store(acc[i] + cval, op + (size_t)i * DOUT);
    }
  }
}

extern "C" void kernel_launch(void* const* d_in, const int* in_sizes, int n_in,
                              void* d_out, int out_size, void* d_ws, size_t ws_size,
                              hipStream_t stream) {
  const float* x = (const float*)d_in[0];   // (B,S,DIN) f32, flat (8192, 4096)
  const float* W = (const float*)d_in[1];   // (DOUT, DIN) f32
  float* out  = (float*)d_out;              // (8192, 4096) f32
  float* c_ws = (float*)d_ws;               // scalar scratch

  const int M = in_sizes[0] / DIN;          // 8192 rows

  et_reduce_c<<<1, 256, 0, stream>>>(x, W, c_ws);

  dim3 grid(DOUT / 512, M / 16);            // 8 x 512 blocks, 256 threads (8 waves)
  et_rank16_wmma<<<grid, 256, 0, stream>>>(x, W, c_ws, out);
}